// ReprogrammingLayer_17626545783527
// MI455X (gfx1250) — compile-verified
//
#include <hip/hip_runtime.h>
#include <math.h>

// Problem constants (from reference: [64,2048,1024] patches, [50257,1024] lexicon, k=5)
#define B_    64
#define L_    2048
#define D_    1024
#define V_    50257
#define K_TOP 5
#define EPS_  1e-8f

#define NCHUNK 16
#define LCHUNK (L_ / NCHUNK)   // 128

#define NTILE 32               // lexicon rows per block
#define KC    128              // K chunk staged in LDS
#define AS_STRIDE 132          // padded (132 mod 64 = 4 -> conflict-free b64 reads)
#define BS_STRIDE 132

typedef __attribute__((ext_vector_type(2))) float v2f;
typedef __attribute__((ext_vector_type(4))) float v4f;
typedef __attribute__((ext_vector_type(8))) float v8f;

// ---------------------------------------------------------------------------
// Kernel 1: partial mean-pool sums. grid (64, 16), block 256.
// Each block: batch b, L-chunk c; thread owns 4 contiguous d (float4 loads).
// Reads the dominant 512MB; fully coalesced, deterministic (no atomics).
// ---------------------------------------------------------------------------
__global__ __launch_bounds__(256) void meanpool_partial(
    const float* __restrict__ pe, float* __restrict__ partial) {
  const int b = blockIdx.x, c = blockIdx.y, t = threadIdx.x;
  const float* src = pe + ((size_t)(b * L_ + c * LCHUNK)) * D_ + t * 4;
  v4f acc = {0.f, 0.f, 0.f, 0.f};
  for (int l = 0; l < LCHUNK; ++l) {
    acc += *(const v4f*)(src + (size_t)l * D_);
  }
  *(v4f*)(partial + ((size_t)(c * B_ + b)) * D_ + t * 4) = acc;
}

// ---------------------------------------------------------------------------
// Kernel 2: reduce 16 partials -> ts (scaled by 1/L), compute x_norm[b].
// grid 64, block 256.
// ---------------------------------------------------------------------------
__global__ __launch_bounds__(256) void finalize_ts(
    const float* __restrict__ partial, float* __restrict__ ts,
    float* __restrict__ xnorm) {
  const int b = blockIdx.x, t = threadIdx.x;
  v4f s = {0.f, 0.f, 0.f, 0.f};
  for (int c = 0; c < NCHUNK; ++c)
    s += *(const v4f*)(partial + ((size_t)(c * B_ + b)) * D_ + t * 4);
  s *= (1.0f / (float)L_);
  *(v4f*)(ts + (size_t)b * D_ + t * 4) = s;

  __shared__ float red[256];
  red[t] = s.x * s.x + s.y * s.y + s.z * s.z + s.w * s.w;
  __syncthreads();
  for (int off = 128; off > 0; off >>= 1) {
    if (t < off) red[t] += red[t + off];
    __syncthreads();
  }
  if (t == 0) xnorm[b] = sqrtf(red[0]);
}

// ---------------------------------------------------------------------------
// Kernel 3: similarity GEMM with fused lexicon row norms + cosine epilogue.
// grid ceil(V/32), block 256 = 8 wave32s.
// Wave w -> (m_tile = w&3, n_tile = w>>2): 4x2 tiles of 16x16 cover M=64,N=32.
// Inner loop: v_wmma_f32_16x16x4_f32, K chunked at 128 via LDS.
// A/B per-lane fragment (ISA 32-bit 16x4 layout): float2 at [row][k + 2*half].
// ---------------------------------------------------------------------------
__global__ __launch_bounds__(256) void gemm_sim(
    const float* __restrict__ ts, const float* __restrict__ lex,
    const float* __restrict__ xnorm, float* __restrict__ sim) {
  __shared__ float As[B_ * AS_STRIDE];        // 33.8 KB
  __shared__ float Bs[NTILE * BS_STRIDE];     // 16.9 KB
  __shared__ float ysq_red[NTILE * 32];       // 4 KB deterministic reduction
  __shared__ float ysq[NTILE];
  __shared__ float xn_s[B_];

  const int t = threadIdx.x;
  const int n0 = blockIdx.x * NTILE;
  const int lane = t & 31, half = lane >> 4, l16 = lane & 15;
  const int wave = t >> 5;
  const int m_tile = wave & 3, n_tile = wave >> 2;
  const int a_row = m_tile * 16 + l16;
  const int b_row = n_tile * 16 + l16;
  const int koff = half * 2;

  if (t < B_) xn_s[t] = xnorm[t];

  float ysq_part[4] = {0.f, 0.f, 0.f, 0.f};   // row (t>>5)+8i, fixed across chunks
  v8f acc = {0.f, 0.f, 0.f, 0.f, 0.f, 0.f, 0.f, 0.f};

  for (int kc = 0; kc < D_; kc += KC) {
    __syncthreads();  // previous chunk's LDS reads done before overwrite
    // Stage A: 64x128 floats, 8 float4 per thread.
#pragma unroll
    for (int i = 0; i < 8; ++i) {
      const int idx = t + i * 256;
      const int row = idx >> 5, c4 = (idx & 31) << 2;
      v4f v = *(const v4f*)(ts + (size_t)row * D_ + kc + c4);
      *(v4f*)(As + row * AS_STRIDE + c4) = v;
    }
    // Stage B: 32x128 floats, 4 float4 per thread; fuse sum-of-squares.
#pragma unroll
    for (int i = 0; i < 4; ++i) {
      const int idx = t + i * 256;
      const int row = idx >> 5, c4 = (idx & 31) << 2;
      int gr = n0 + row; if (gr > V_ - 1) gr = V_ - 1;   // clamp tail block
      v4f v = *(const v4f*)(lex + (size_t)gr * D_ + kc + c4);
      *(v4f*)(Bs + row * BS_STRIDE + c4) = v;
      ysq_part[i] += v.x * v.x + v.y * v.y + v.z * v.z + v.w * v.w;
    }
    __syncthreads();
    const float* ap = As + a_row * AS_STRIDE + koff;
    const float* bp = Bs + b_row * BS_STRIDE + koff;
#pragma unroll
    for (int k = 0; k < KC; k += 4) {
      v2f a = *(const v2f*)(ap + k);
      v2f b = *(const v2f*)(bp + k);
      acc = __builtin_amdgcn_wmma_f32_16x16x4_f32(
          /*neg_a=*/false, a, /*neg_b=*/false, b,
          /*c_mod=*/(short)0, acc, /*reuse_a=*/false, /*reuse_b=*/false);
    }
  }

  // Deterministic y-norm reduction: 32 partials per row, fixed order.
#pragma unroll
  for (int i = 0; i < 4; ++i) {
    const int row = (t >> 5) + 8 * i;
    ysq_red[row * 32 + (t & 31)] = ysq_part[i];
  }
  __syncthreads();
  if (t < NTILE) {
    float s = 0.f;
    for (int j = 0; j < 32; ++j) s += ysq_red[t * 32 + j];
    ysq[t] = s;
  }
  __syncthreads();

  // Epilogue: cosine normalize and store. C layout: VGPR j -> M = j + 8*half.
  const int n_local = n_tile * 16 + l16;
  const int n_global = n0 + n_local;
  if (n_global < V_) {
    const float yn = sqrtf(ysq[n_local]);
#pragma unroll
    for (int j = 0; j < 8; ++j) {
      const int mg = m_tile * 16 + half * 8 + j;
      const float denom = fmaxf(xn_s[mg] * yn, EPS_);
      sim[(size_t)mg * V_ + n_global] = acc[j] / denom;
    }
  }
}

// ---------------------------------------------------------------------------
// Kernel 4: per-row top-5 + gather lexicon rows. grid 64, block 256.
// ---------------------------------------------------------------------------
__global__ __launch_bounds__(256) void topk_gather(
    const float* __restrict__ sim, const float* __restrict__ lex,
    float* __restrict__ out_topk) {
  const int b = blockIdx.x, t = threadIdx.x;
  const float* row = sim + (size_t)b * V_;

  float bv[K_TOP]; int bi[K_TOP];
#pragma unroll
  for (int j = 0; j < K_TOP; ++j) { bv[j] = -3.4e38f; bi[j] = 0x7fffffff; }

  for (int n = t; n < V_; n += 256) {
    const float s = row[n];
    if (s > bv[K_TOP - 1]) {
      bv[K_TOP - 1] = s; bi[K_TOP - 1] = n;
#pragma unroll
      for (int j = K_TOP - 1; j > 0; --j) {
        if (bv[j] > bv[j - 1]) {
          float tv = bv[j]; bv[j] = bv[j - 1]; bv[j - 1] = tv;
          int ti = bi[j]; bi[j] = bi[j - 1]; bi[j - 1] = ti;
        }
      }
    }
  }

  __shared__ float sv[256 * K_TOP];
  __shared__ int   si[256 * K_TOP];
  __shared__ int   sel[K_TOP];
#pragma unroll
  for (int j = 0; j < K_TOP; ++j) { sv[t * K_TOP + j] = bv[j]; si[t * K_TOP + j] = bi[j]; }
  __syncthreads();

  if (t == 0) {  // serial merge of 1280 candidates; ties -> smallest index
    for (int r = 0; r < K_TOP; ++r) {
      float best = -3.4e38f; int bidx = 0x7fffffff; int bslot = 0;
      for (int i = 0; i < 256 * K_TOP; ++i) {
        const float v = sv[i];
        if (v > best || (v == best && si[i] < bidx)) { best = v; bidx = si[i]; bslot = i; }
      }
      sel[r] = bidx;
      sv[bslot] = -3.4e38f;
    }
  }
  __syncthreads();

#pragma unroll
  for (int r = 0; r < K_TOP; ++r) {
    const int src = sel[r];
    v4f v = *(const v4f*)(lex + (size_t)src * D_ + t * 4);
    *(v4f*)(out_topk + ((size_t)(b * K_TOP + r)) * D_ + t * 4) = v;
  }
}

// ---------------------------------------------------------------------------
extern "C" void kernel_launch(void* const* d_in, const int* in_sizes, int n_in,
                              void* d_out, int out_size, void* d_ws, size_t ws_size,
                              hipStream_t stream) {
  const float* pe  = (const float*)d_in[0];   // [64, 2048, 1024] f32
  const float* lex = (const float*)d_in[1];   // [50257, 1024] f32
  (void)in_sizes; (void)n_in; (void)out_size; (void)ws_size;  // k==5 baked in

  float* ws      = (float*)d_ws;
  float* partial = ws;                                  // NCHUNK*B*D floats (4 MB)
  float* ts      = ws + (size_t)NCHUNK * B_ * D_;       // B*D floats
  float* xnorm   = ts + (size_t)B_ * D_;                // B floats

  float* out_topk = (float*)d_out;                      // [64, 5, 1024]
  float* sim      = out_topk + (size_t)B_ * K_TOP * D_; // [64, 50257]

  dim3 g1(B_, NCHUNK);
  meanpool_partial<<<g1, 256, 0, stream>>>(pe, partial);
  finalize_ts<<<B_, 256, 0, stream>>>(partial, ts, xnorm);

  const int nblk = (V_ + NTILE - 1) / NTILE;   // 1571
  gemm_sim<<<nblk, 256, 0, stream>>>(ts, lex, xnorm, sim);

  topk_gather<<<B_, 256, 0, stream>>>(sim, lex, out_topk);
}